// GAT_3753801416709
// MI455X (gfx1250) — compile-verified
//
#include <hip/hip_runtime.h>

// ---------------------------------------------------------------------------
// GAT pipeline for MI455X (gfx1250, wave32). Dense per-head linears run on
// V_WMMA_F32_16X16X4_F32 (full f32 precision, matching the f32 reference);
// edge aggregation is the memory-bound part and uses global f32 atomics into
// an L2-resident [N,256] accumulator. conv5/conv6/bn5/bn6 are dead code in
// the reference dataflow (x6 is never consumed) and are skipped.
// ---------------------------------------------------------------------------

#define N_NODES 20000
#define N_EDGES 320000
#define HEADS   4
#define CH      256         // HEADS * 64
#define BSZ     40
#define NPG     500         // N_NODES / BSZ
#define GAT_EPS 1e-5f

typedef float v2f __attribute__((ext_vector_type(2)));
typedef float v8f __attribute__((ext_vector_type(8)));

__device__ __forceinline__ void atomic_add_f32(float* p, float v) {
  __hip_atomic_fetch_add(p, v, __ATOMIC_RELAXED, __HIP_MEMORY_SCOPE_AGENT);
}

// -------------------------------- utility ----------------------------------
__global__ void k_zero(float* __restrict__ p, int n) {
  int i = blockIdx.x * blockDim.x + threadIdx.x;
  int stride = gridDim.x * blockDim.x;
  for (; i < n; i += stride) p[i] = 0.f;
}

__global__ void k_invsq(const float* __restrict__ ea, float* __restrict__ isq, int n) {
  int i = blockIdx.x * blockDim.x + threadIdx.x;
  if (i < n) {
    float t = ea[i] + 1e-7f;
    isq[i] = 1.f / (t * t);
  }
}

// ------------------------- per-head linear (WMMA) --------------------------
// XH[h][n][o] = sum_f X[n][f] * W[h][f][o] + B[h][o]
// One block = 16 rows of X (staged in LDS), 8 waves x 2 N-tiles = all 256 cols.
template <int FIN>
__global__ void k_gemm_wmma(const float* __restrict__ X, const float* __restrict__ W,
                            const float* __restrict__ Bb, float* __restrict__ XH) {
  __shared__ float Xs[16][FIN + 1];  // +1 pad: conflict-free column reads
  const int m0  = blockIdx.x * 16;
  const int tid = threadIdx.x;
  for (int i = tid; i < 16 * FIN; i += 256) {
    int r = i / FIN, c = i - r * FIN;
    Xs[r][c] = X[(size_t)(m0 + r) * FIN + c];
  }
  __syncthreads();

  const int lane = tid & 31;
  const int wid  = tid >> 5;
  const int row  = lane & 15;        // M (for A) / N (for B) within tile
  const int ksel = (lane >> 4) * 2;  // lanes 0-15: K+0/1, lanes 16-31: K+2/3

  #pragma unroll
  for (int j = 0; j < 2; ++j) {
    const int nt = wid * 2 + j;      // 16 N-tiles of 16 cols
    const int c0 = nt * 16;
    const int h  = c0 >> 6;
    const int o  = c0 & 63;
    const float* Wh = W + (size_t)h * FIN * 64 + o + row;

    v8f acc = {};
    for (int k0 = 0; k0 < FIN; k0 += 4) {
      v2f a, b;
      a.x = Xs[row][k0 + ksel];
      a.y = Xs[row][k0 + ksel + 1];
      b.x = Wh[(size_t)(k0 + ksel) * 64];
      b.y = Wh[(size_t)(k0 + ksel + 1) * 64];
      acc = __builtin_amdgcn_wmma_f32_16x16x4_f32(
          /*neg_a=*/false, a, /*neg_b=*/false, b,
          /*c_mod=*/(short)0, acc, /*reuse_a=*/false, /*reuse_b=*/false);
    }

    const float bias  = Bb[h * 64 + o + row];          // bias depends on col only
    const int   rbase = m0 + ((lane >> 4) << 3);       // lanes>=16 hold rows v+8
    float* outp = XH + ((size_t)h * N_NODES) * 64 + o + row;
    #pragma unroll
    for (int v = 0; v < 8; ++v)
      outp[(size_t)(rbase + v) * 64] = acc[v] + bias;
  }
}

// ---------------------- attention logits a_i / a_j -------------------------
// ai[h*N+n] = dot(XH[h][n][:], att[h][0:64]); aj uses att[h][64:128].
__global__ void k_aiaj(const float* __restrict__ XH, const float* __restrict__ att,
                       float* __restrict__ ai, float* __restrict__ aj) {
  const int gw   = (blockIdx.x * blockDim.x + threadIdx.x) >> 5;  // = h*N + n
  const int lane = threadIdx.x & 31;
  if (gw >= HEADS * N_NODES) return;
  const int h = gw / N_NODES;
  const float* rowp = XH + (size_t)gw * 64;
  const float* at   = att + h * 128;
  float x0 = rowp[lane], x1 = rowp[lane + 32];
  float si = x0 * at[lane]      + x1 * at[lane + 32];
  float sj = x0 * at[64 + lane] + x1 * at[96 + lane];
  #pragma unroll
  for (int off = 16; off > 0; off >>= 1) {
    si += __shfl_xor(si, off, 32);
    sj += __shfl_xor(sj, off, 32);
  }
  if (lane == 0) { ai[gw] = si; aj[gw] = sj; }
}

// --------------------------- edge aggregation ------------------------------
// agg[dst][h*64+o] += leaky(ai[h,dst]+aj[h,src]) * inv_sq[e] * XH[h][src][o]
__global__ void k_edge(const int* __restrict__ src, const int* __restrict__ dst,
                       const float* __restrict__ isq, const float* __restrict__ ai,
                       const float* __restrict__ aj, const float* __restrict__ XH,
                       float* __restrict__ agg) {
  const int e    = (blockIdx.x * blockDim.x + threadIdx.x) >> 5;
  const int lane = threadIdx.x & 31;
  if (e >= N_EDGES) return;
  const int s = src[e], d = dst[e];
  const float w = isq[e];
  float* outp = agg + (size_t)d * CH;
  #pragma unroll
  for (int h = 0; h < HEADS; ++h) {
    float sc = ai[h * N_NODES + d] + aj[h * N_NODES + s];
    sc = sc > 0.f ? sc : 0.01f * sc;   // leaky_relu, slope 0.01
    sc *= w;
    const float* rowp = XH + ((size_t)h * N_NODES + s) * 64;
    atomic_add_f32(outp + h * 64 + lane,      sc * rowp[lane]);
    atomic_add_f32(outp + h * 64 + lane + 32, sc * rowp[lane + 32]);
  }
}

// ------------------------------- batch norm --------------------------------
__global__ void k_bnstats(const float* __restrict__ Xc, float* __restrict__ stats) {
  const int c     = threadIdx.x;                 // 256 channels
  const int chunk = N_NODES / gridDim.x;         // launched with grid=100 -> 200 rows
  const int r0    = blockIdx.x * chunk;
  float s = 0.f, s2 = 0.f;
  for (int r = r0; r < r0 + chunk; ++r) {
    float v = Xc[(size_t)r * CH + c];
    s += v; s2 += v * v;
  }
  atomic_add_f32(&stats[c], s);
  atomic_add_f32(&stats[CH + c], s2);
}

__global__ void k_bnapply(float* __restrict__ Xc, const float* __restrict__ stats,
                          const float* __restrict__ g, const float* __restrict__ b) {
  const int i = blockIdx.x * blockDim.x + threadIdx.x;
  if (i >= N_NODES * CH) return;
  const int c = i & (CH - 1);
  const float invN = 1.f / (float)N_NODES;
  float m   = stats[c] * invN;
  float var = stats[CH + c] * invN - m * m;      // biased variance (torch BN norm)
  float y = g[c] * (Xc[i] - m) * rsqrtf(var + GAT_EPS) + b[c];
  Xc[i] = y > 0.f ? y : 0.f;                     // fused ReLU
}

// ---------------------------- head-mean (conv_f) ---------------------------
__global__ void k_headmean(const float* __restrict__ agg, float* __restrict__ xf) {
  const int i = blockIdx.x * blockDim.x + threadIdx.x;  // N*64
  if (i >= N_NODES * 64) return;
  const int n = i >> 6, o = i & 63;
  const float* p = agg + (size_t)n * CH + o;
  xf[i] = 0.25f * (p[0] + p[64] + p[128] + p[192]);
}

// ------------------------------- MLP head ----------------------------------
__global__ void k_head(const float* __restrict__ xf, const int* __restrict__ lini,
                       const float* bg1, const float* bb1, const float* bg2, const float* bb2,
                       const float* bg3, const float* bb3, const float* bg4, const float* bb4,
                       const float* W1, const float* B1, const float* W2, const float* B2,
                       const float* W3, const float* B3, const float* W4, const float* B4,
                       const float* Wp, const float* Bp, float* __restrict__ out) {
  __shared__ float smem[3 * BSZ * 128 + BSZ * 64];
  float* H  = smem;
  float* H2 = smem + BSZ * 128;
  float* T  = smem + 2 * BSZ * 128;
  float* F  = smem + 3 * BSZ * 128;
  const int tid = threadIdx.x;

  // feat = 0.5*(xf[li] + xf[ni])
  for (int i = tid; i < BSZ * 64; i += 256) {
    int r = i >> 6, c = i & 63;
    int li = lini[2 * r]     + r * NPG;
    int ni = lini[2 * r + 1] + r * NPG;
    F[i] = 0.5f * (xf[li * 64 + c] + xf[ni * 64 + c]);
  }
  __syncthreads();

  // T = relu(bn(F)) over 40 rows, dim 64
  if (tid < 64) {
    float m = 0.f;
    for (int r = 0; r < BSZ; ++r) m += F[r * 64 + tid];
    m *= (1.f / BSZ);
    float v = 0.f;
    for (int r = 0; r < BSZ; ++r) { float d = F[r * 64 + tid] - m; v += d * d; }
    v *= (1.f / BSZ);
    float sc = bg1[tid] * rsqrtf(v + GAT_EPS);
    for (int r = 0; r < BSZ; ++r) {
      float y = sc * (F[r * 64 + tid] - m) + bb1[tid];
      T[r * 64 + tid] = y > 0.f ? y : 0.f;
    }
  }
  __syncthreads();

  // H = T(40x64) @ W1(64x128) + B1
  for (int i = tid; i < BSZ * 128; i += 256) {
    int r = i >> 7, c = i & 127;
    float s = B1[c];
    for (int k = 0; k < 64; ++k) s += T[r * 64 + k] * W1[k * 128 + c];
    H[i] = s;
  }
  __syncthreads();

  const float* bgs[3] = {bg2, bg3, bg4};
  const float* bbs[3] = {bb2, bb3, bb4};
  const float* Ws[3]  = {W2, W3, W4};
  const float* Bs[3]  = {B2, B3, B4};
  for (int L = 0; L < 3; ++L) {
    if (tid < 128) {
      float m = 0.f;
      for (int r = 0; r < BSZ; ++r) m += H[r * 128 + tid];
      m *= (1.f / BSZ);
      float v = 0.f;
      for (int r = 0; r < BSZ; ++r) { float d = H[r * 128 + tid] - m; v += d * d; }
      v *= (1.f / BSZ);
      float sc = bgs[L][tid] * rsqrtf(v + GAT_EPS);
      for (int r = 0; r < BSZ; ++r) {
        float y = sc * (H[r * 128 + tid] - m) + bbs[L][tid];
        T[r * 128 + tid] = y > 0.f ? y : 0.f;
      }
    }
    __syncthreads();
    for (int i = tid; i < BSZ * 128; i += 256) {
      int r = i >> 7, c = i & 127;
      float s = Bs[L][c] + H[i];                 // residual
      for (int k = 0; k < 128; ++k) s += T[r * 128 + k] * Ws[L][k * 128 + c];
      H2[i] = s;
    }
    __syncthreads();
    float* tmp = H; H = H2; H2 = tmp;
  }

  if (tid < BSZ) {
    float s = Bp[0];
    for (int k = 0; k < 128; ++k) s += H[tid * 128 + k] * Wp[k];
    out[tid] = s;
  }
}

// ------------------------------- launcher ----------------------------------
extern "C" void kernel_launch(void* const* d_in, const int* in_sizes, int n_in,
                              void* d_out, int out_size, void* d_ws, size_t ws_size,
                              hipStream_t stream) {
  (void)in_sizes; (void)n_in; (void)out_size; (void)ws_size;

  // setup_inputs() insertion-order, depth-first flattening:
  // 0:x 1:edge_attr
  // params: conv1..conv6,conv_f (W,b,att) @2..22 ; bn1..bn6 (g,b) @23..34 ;
  //         bnh1..bnh4 (g,b) @35..42 ; h1..h4,pred (W,b) @43..52
  // 53:edge_index 54:batch 55:li_ni 56:k_nb 57:u_nb 58:v_nb 59:num_nodes
  auto P = [&](int i) { return (const float*)d_in[i]; };
  const float* x  = P(0);
  const float* ea = P(1);
  const float* cW[5] = {P(2), P(5), P(8),  P(11), P(20)};  // conv1,2,3,4,conv_f
  const float* cB[5] = {P(3), P(6), P(9),  P(12), P(21)};
  const float* cA[5] = {P(4), P(7), P(10), P(13), P(22)};
  const float* bnG[4] = {P(23), P(25), P(27), P(29)};      // bn1..bn4
  const float* bnB[4] = {P(24), P(26), P(28), P(30)};
  const int* ei   = (const int*)d_in[53];
  const int* lini = (const int*)d_in[55];
  const int* src = ei;
  const int* dst = ei + N_EDGES;

  // workspace layout (floats); ~109 MB total
  float* ws = (float*)d_ws;
  float* xh = ws;                 // [4][20000][64]  = 5,120,000
  float* x1 = xh + 5120000;       // [20000][256]
  float* x2 = x1 + 5120000;
  float* x3 = x2 + 5120000;
  float* x4 = x3 + 5120000;
  float* xf = x4 + 5120000;       // [20000][64]     = 1,280,000
  float* ai = xf + 1280000;       // [4][20000]
  float* aj = ai + 80000;
  float* isq = aj + 80000;        // [320000]
  float* stats = isq + 320000;    // [512]

  dim3 blk(256);
  k_invsq<<<(N_EDGES + 255) / 256, blk, 0, stream>>>(ea, isq, N_EDGES);

  auto conv = [&](const float* Xin, int fin, int ci, float* agg, bool zero_agg) {
    if (fin == 64)
      k_gemm_wmma<64><<<N_NODES / 16, blk, 0, stream>>>(Xin, cW[ci], cB[ci], xh);
    else
      k_gemm_wmma<256><<<N_NODES / 16, blk, 0, stream>>>(Xin, cW[ci], cB[ci], xh);
    k_aiaj<<<(HEADS * N_NODES) / 8, blk, 0, stream>>>(xh, cA[ci], ai, aj);
    if (zero_agg) k_zero<<<2048, blk, 0, stream>>>(agg, N_NODES * CH);
    k_edge<<<N_EDGES / 8, blk, 0, stream>>>(src, dst, isq, ai, aj, xh, agg);
  };
  auto bn = [&](float* agg, int bi) {
    k_zero<<<2, blk, 0, stream>>>(stats, 2 * CH);
    k_bnstats<<<100, blk, 0, stream>>>(agg, stats);
    k_bnapply<<<(N_NODES * CH + 255) / 256, blk, 0, stream>>>(agg, stats, bnG[bi], bnB[bi]);
  };

  conv(x,  64,  0, x1, true);  bn(x1, 0);   // x1 = relu(bn1(conv1(x)))
  conv(x1, 256, 1, x2, true);  bn(x2, 1);   // x2
  conv(x2, 256, 2, x3, true);               // conv3(x2)
  conv(x1, 256, 2, x3, false);              // + conv3(x1)  (atomic accumulate)
  bn(x3, 2);                                // x3
  conv(x3, 256, 3, x4, true);  bn(x4, 3);   // x4
  // conv5/conv6/bn5/bn6 are dead code: x6 never feeds the output.
  conv(x4, 256, 4, x1, true);               // conv_f -> agg reuses x1
  k_headmean<<<(N_NODES * 64 + 255) / 256, blk, 0, stream>>>(x1, xf);

  k_head<<<1, blk, 0, stream>>>(xf, lini,
      P(35), P(36), P(37), P(38), P(39), P(40), P(41), P(42),
      P(43), P(44), P(45), P(46), P(47), P(48), P(49), P(50),
      P(51), P(52), (float*)d_out);
}